// SpatioTemporalBlock_77378130805060
// MI455X (gfx1250) — compile-verified
//
#include <hip/hip_runtime.h>

// ---------------------------------------------------------------------------
// SpatioTemporalBlock for MI455X (gfx1250, wave32, WMMA).
// out = deform_conv2d(x, offsets(x)) + conv3d_3x3x3(x) + bias
// GEMMs on V_WMMA_F32_16X16X4_F32 (fp32-exact); weight tiles staged with the
// gfx1250 async global->LDS path (ASYNCcnt); all A-tile sampling (im2col and
// deformable bilinear) is branchless: clamped addresses + select-zero masks.
// ---------------------------------------------------------------------------

typedef __attribute__((ext_vector_type(2))) float v2f;
typedef __attribute__((ext_vector_type(8))) float v8f;

#define T_DIM  4
#define H_DIM  48
#define W_DIM  48
#define HW_SZ  2304            // 48*48
#define NPOS   9216            // 4*48*48
#define OFFC   200             // DG*2*25
#define KDEF   6400            // 256*25
#define KTMP   6912            // 256*27

// ---------------------------------------------------------------------------
// coords kernel: offsets [t][200][h][w] -> per (t,p,g,k):
//   wts  = float4  bilinear weights, pre-multiplied by per-corner validity
//   offs = ushort4 clamped pixel offsets (y*48+x) for the 4 corners
// laid out [(p*4+g)*25 + k]
// ---------------------------------------------------------------------------
__global__ __launch_bounds__(256) void coords_kernel(
    const float* __restrict__ off,
    float4*      __restrict__ wts,
    ushort4*     __restrict__ offs)
{
    int idx = blockIdx.x * 256 + threadIdx.x;          // NPOS*4*25 = 921600
    if (idx >= NPOS * 100) return;
    int r  = idx % 25;
    int g  = (idx / 25) & 3;
    int p  = idx / 100;
    int t  = p / HW_SZ;
    int hw = p % HW_SZ;
    int h  = hw / W_DIM;
    int w  = hw % W_DIM;
    float dy = off[(size_t)(t * OFFC + g * 50 + 2 * r    ) * HW_SZ + hw];
    float dx = off[(size_t)(t * OFFC + g * 50 + 2 * r + 1) * HW_SZ + hw];
    float sy = (float)(h + r / 5 - 2) + dy;
    float sx = (float)(w + r % 5 - 2) + dx;
    float fy = floorf(sy), fx = floorf(sx);
    float wy = sy - fy,    wx = sx - fx;
    int y0 = (int)fy, x0 = (int)fx;
    int y1 = y0 + 1,  x1 = x0 + 1;
    float my0 = ((unsigned)y0 < H_DIM) ? 1.f : 0.f;
    float my1 = ((unsigned)y1 < H_DIM) ? 1.f : 0.f;
    float mx0 = ((unsigned)x0 < W_DIM) ? 1.f : 0.f;
    float mx1 = ((unsigned)x1 < W_DIM) ? 1.f : 0.f;
    int y0c = min(max(y0, 0), H_DIM - 1), y1c = min(max(y1, 0), H_DIM - 1);
    int x0c = min(max(x0, 0), W_DIM - 1), x1c = min(max(x1, 0), W_DIM - 1);
    wts[idx] = make_float4((1.f - wy) * (1.f - wx) * my0 * mx0,
                           (1.f - wy) *        wx  * my0 * mx1,
                                  wy  * (1.f - wx) * my1 * mx0,
                                  wy  *        wx  * my1 * mx1);
    offs[idx] = make_ushort4((unsigned short)(y0c * W_DIM + x0c),
                             (unsigned short)(y0c * W_DIM + x1c),
                             (unsigned short)(y1c * W_DIM + x0c),
                             (unsigned short)(y1c * W_DIM + x1c));
}

// ---------------------------------------------------------------------------
// Implicit GEMM: block tile 32(M=positions) x 256(N=channels), Kc=16.
// 256 threads = 8 waves in 2(M) x 4(N); each wave -> four 16x16 f32 C tiles.
// B tile (weights, [Cout][KTOT] row-major) staged via async global->LDS.
// MODE 0: A = 5x5 im2col         -> offsets buffer (N valid = 200)
// MODE 1: A = 3x3x3 im2col       -> out  = acc + bias
// MODE 2: A = bilinear sampled   -> out += acc
// ---------------------------------------------------------------------------
template <int MODE>
__global__ __launch_bounds__(256) void gemm_kernel(
    const float*   __restrict__ X,
    const float*   __restrict__ Wt,
    const float*   __restrict__ bias,
    const float4*  __restrict__ wts,
    const ushort4* __restrict__ offs,
    float*         __restrict__ out)
{
    constexpr int KTOT = (MODE == 1) ? KTMP : KDEF;
    constexpr int RMOD = (MODE == 1) ? 27 : 25;
    constexpr int NVAL = (MODE == 0) ? OFFC : 256;

    __shared__ __align__(16) float As[32][20];    // [m][k], padded stride
    __shared__ __align__(16) float Bs[256][20];   // [n][k], padded stride

    const int tid  = threadIdx.x;
    const int lane = tid & 31;
    const int wave = tid >> 5;
    const int wm   = wave & 1;     // 0..1  M sub-tile
    const int wn   = wave >> 1;    // 0..3  N sub-tile (64 cols each)
    const int hh   = lane >> 4;    // half-wave
    const int col  = lane & 15;

    const int pBase = blockIdx.x * 32;     // 288 blocks cover 9216 positions

    // A-fill mapping: thread fills As[ia][k0], As[ia][k0+1]
    const int ia = tid >> 3;               // 0..31
    const int k0 = (tid & 7) << 1;         // 0,2,..,14
    const int pA  = pBase + ia;
    const int tA  = pA / HW_SZ;
    const int hwA = pA % HW_SZ;
    const int hA  = hwA / W_DIM;
    const int wA  = hwA % W_DIM;

    // incremental K decode: kg = kk*16 + k0 + s  ->  (c, r) with r = kg % RMOD
    int cA = 0, rA = k0;                   // k0 < 16 <= RMOD

    // async B staging: thread owns weight row n = tid (clamped for MODE 0)
    const int gn = (tid < NVAL) ? tid : (NVAL - 1);
    const float* gB = Wt + (size_t)gn * KTOT;
    const unsigned ldsB = (unsigned)(uintptr_t)(&Bs[tid][0]);

    v8f acc[4] = {};

    for (int kk = 0; kk < KTOT / 16; ++kk) {
        // ---- stage A tile (branchless im2col / bilinear sample) ----
        #pragma unroll
        for (int s = 0; s < 2; ++s) {
            int rs = rA + s, cs = cA;
            if (rs >= RMOD) { rs -= RMOD; cs += 1; }
            float v;
            if (MODE == 0) {                       // 5x5 im2col, clamped+masked
                int yy = hA + rs / 5 - 2;
                int xx = wA + rs % 5 - 2;
                bool ok = ((unsigned)yy < H_DIM) & ((unsigned)xx < W_DIM);
                int yyc = min(max(yy, 0), H_DIM - 1);
                int xxc = min(max(xx, 0), W_DIM - 1);
                float xv = X[(size_t)(cs * T_DIM + tA) * HW_SZ
                             + yyc * W_DIM + xxc];
                v = ok ? xv : 0.0f;
            } else if (MODE == 1) {                // 3x3x3 im2col, clamped+masked
                int dt = rs / 9;
                int r9 = rs - dt * 9;
                int tt = tA + dt - 1;
                int yy = hA + r9 / 3 - 1;
                int xx = wA + r9 % 3 - 1;
                bool ok = ((unsigned)tt < T_DIM) & ((unsigned)yy < H_DIM) &
                          ((unsigned)xx < W_DIM);
                int ttc = min(max(tt, 0), T_DIM - 1);
                int yyc = min(max(yy, 0), H_DIM - 1);
                int xxc = min(max(xx, 0), W_DIM - 1);
                float xv = X[(size_t)(cs * T_DIM + ttc) * HW_SZ
                             + yyc * W_DIM + xxc];
                v = ok ? xv : 0.0f;
            } else {                               // branchless bilinear
                int g = cs >> 6;                   // deform group (Cg = 64)
                size_t ci = (size_t)(pA * 4 + g) * 25 + rs;
                float4  wv = wts[ci];
                ushort4 ov = offs[ci];
                const float* img = X + (size_t)(cs * T_DIM + tA) * HW_SZ;
                v = wv.x * img[ov.x] + wv.y * img[ov.y]
                  + wv.z * img[ov.z] + wv.w * img[ov.w];
            }
            As[ia][k0 + s] = v;
        }
        rA += 16;
        if (rA >= RMOD) { rA -= RMOD; cA += 1; }

        // ---- stage B tile: 4 x 16B async global->LDS per thread ----
        {
            const float* gsrc = gB + kk * 16;
            #pragma unroll
            for (int q = 0; q < 4; ++q) {
                asm volatile("global_load_async_to_lds_b128 %0, %1, off"
                             :: "v"(ldsB + (unsigned)(q * 16)),
                                "v"(gsrc + q * 4)
                             : "memory");
            }
        }
        asm volatile("s_wait_asynccnt 0" ::: "memory");
        __syncthreads();

        // ---- 4 WMMA k-steps of K=4, 4 N-tiles per wave ----
        #pragma unroll
        for (int ks = 0; ks < 4; ++ks) {
            const int kb = ks * 4 + hh * 2;
            v2f a;
            a[0] = As[wm * 16 + col][kb];
            a[1] = As[wm * 16 + col][kb + 1];
            #pragma unroll
            for (int j = 0; j < 4; ++j) {
                const int n = wn * 64 + j * 16 + col;
                v2f b;
                b[0] = Bs[n][kb];
                b[1] = Bs[n][kb + 1];
                acc[j] = __builtin_amdgcn_wmma_f32_16x16x4_f32(
                             false, a, false, b, (short)0, acc[j],
                             false, false);
            }
        }

        __syncthreads();
    }

    // ---- store: acc[j][r] holds C[M = r + 8*hh][N = col] of tile j ----
    #pragma unroll
    for (int j = 0; j < 4; ++j) {
        const int n = wn * 64 + j * 16 + col;
        #pragma unroll
        for (int r = 0; r < 8; ++r) {
            const int p = pBase + wm * 16 + hh * 8 + r;
            if (MODE == 0) {
                if (n < OFFC) {
                    int t  = p / HW_SZ;
                    int hw = p % HW_SZ;
                    out[(size_t)(t * OFFC + n) * HW_SZ + hw] = acc[j][r];
                }
            } else if (MODE == 1) {
                out[(size_t)n * NPOS + p] = acc[j][r] + bias[n];
            } else {
                out[(size_t)n * NPOS + p] += acc[j][r];
            }
        }
    }
}

// ---------------------------------------------------------------------------
extern "C" void kernel_launch(void* const* d_in, const int* in_sizes, int n_in,
                              void* d_out, int out_size, void* d_ws, size_t ws_size,
                              hipStream_t stream)
{
    (void)in_sizes; (void)n_in; (void)out_size; (void)ws_size;

    const float* x      = (const float*)d_in[0];   // [1,256,4,48,48]
    const float* w_off  = (const float*)d_in[1];   // [200,256,1,5,5]
    const float* w_def  = (const float*)d_in[2];   // [256,256,5,5]
    const float* w_temp = (const float*)d_in[3];   // [256,256,3,3,3]
    const float* b_temp = (const float*)d_in[4];   // [256]
    float* out = (float*)d_out;                    // [1,256,4,48,48]

    // workspace: offsets [4][200][48][48] f32 (7.37 MB),
    //            wts float4 (14.75 MB), offs ushort4 (7.37 MB)
    char* ws = (char*)d_ws;
    float*   off  = (float*)ws;
    float4*  wts  = (float4*)(ws + (size_t)T_DIM * OFFC * HW_SZ * 4);
    ushort4* offs = (ushort4*)(ws + (size_t)T_DIM * OFFC * HW_SZ * 4
                                  + (size_t)NPOS * 100 * sizeof(float4));

    dim3 blk(256);

    // 1) offset conv (im2col GEMM, N valid = 200)
    gemm_kernel<0><<<dim3(NPOS / 32, 1), blk, 0, stream>>>(
        x, w_off, nullptr, nullptr, nullptr, off);

    // 2) bilinear tap weights/offsets per (t, pos, group, tap)
    coords_kernel<<<dim3((NPOS * 100 + 255) / 256), blk, 0, stream>>>(
        off, wts, offs);

    // 3) temporal 3x3x3 conv GEMM: out = acc + bias
    gemm_kernel<1><<<dim3(NPOS / 32, 1), blk, 0, stream>>>(
        x, w_temp, b_temp, nullptr, nullptr, out);

    // 4) deformable conv GEMM: out += acc
    gemm_kernel<2><<<dim3(NPOS / 32, 1), blk, 0, stream>>>(
        x, w_def, nullptr, wts, offs, out);
}